// GraphLevelGCN_58171037057469
// MI455X (gfx1250) — compile-verified
//
#include <hip/hip_runtime.h>
#include <hip/hip_bf16.h>

// ---------------------------------------------------------------------------
// GCN: h1 = relu(Norm(x@W1)+b1); h2 = Norm(h1@W2)+b2; out = segment_mean(h2)
// Norm(h)[i] = sum_{e:dst=i} dinv[src]*dinv[dst]*h[src]  (+ self loop i->i)
// N=50000, E=800000, D=H=128, G=64.
// ---------------------------------------------------------------------------

typedef __attribute__((ext_vector_type(2))) float v2f;
typedef __attribute__((ext_vector_type(8))) float v8f;

#define GCN_H 128

// ---------- small utility kernels ----------
__global__ void gcn_zero(float* __restrict__ p, int n) {
    int i = blockIdx.x * blockDim.x + threadIdx.x;
    if (i < n) p[i] = 0.0f;
}

__global__ void gcn_deg_init(float* __restrict__ deg, int n) {
    int i = blockIdx.x * blockDim.x + threadIdx.x;
    if (i < n) deg[i] = 1.0f;   // self loop contributes 1 to every node's degree
}

__global__ void gcn_deg_count(const int* __restrict__ dst, float* __restrict__ deg, int E) {
    int e = blockIdx.x * blockDim.x + threadIdx.x;
    if (e < E) atomicAdd(&deg[dst[e]], 1.0f);
}

__global__ void gcn_dinv(const float* __restrict__ deg, float* __restrict__ dinv, int n) {
    int i = blockIdx.x * blockDim.x + threadIdx.x;
    if (i < n) {
        float d = deg[i];
        dinv[i] = (d > 0.0f) ? rsqrtf(d) : 0.0f;
    }
}

// ---------- f32 WMMA GEMM: Y[M x 128] = X[M x 128] @ W[128 x 128] ----------
// One block = 256 threads = 8 waves. Block handles rows [row0, row0+16),
// wave w handles output columns [16w, 16w+16). K=128 via 32 x wmma 16x16x4.
__global__ __launch_bounds__(256) void gcn_gemm_f32(const float* __restrict__ X,
                                                    const float* __restrict__ W,
                                                    float* __restrict__ Y) {
    __shared__ float xs[16][132];          // +4 pad: stride 132 % 64 banks = 4
    const int tid  = threadIdx.x;
    const int row0 = blockIdx.x * 16;

    // Stage 16x128 tile of X into LDS (coalesced, 8 floats per thread).
    for (int idx = tid; idx < 16 * GCN_H; idx += 256) {
        int r = idx >> 7, c = idx & 127;
        xs[r][c] = X[(size_t)(row0 + r) * GCN_H + c];
    }
    __syncthreads();

    const int wave = tid >> 5;
    const int lane = tid & 31;
    const int half = lane >> 4;            // 0: K pair {0,1}, 1: K pair {2,3}
    const int m    = lane & 15;            // A: M index / B: N index
    const int col0 = wave * 16;

    v8f acc = {};
#pragma unroll
    for (int kk = 0; kk < 32; ++kk) {
        const int kb = kk * 4 + half * 2;
        // A fragment: lane holds X[row0+m][kb], X[row0+m][kb+1]
        v2f a;
        a.x = xs[m][kb];
        a.y = xs[m][kb + 1];
        // B fragment: lane holds W[kb][col0+m], W[kb+1][col0+m]
        v2f b;
        b.x = W[(size_t)kb * GCN_H + col0 + m];
        b.y = W[(size_t)(kb + 1) * GCN_H + col0 + m];
        acc = __builtin_amdgcn_wmma_f32_16x16x4_f32(false, a, false, b,
                                                    (short)0, acc, false, false);
    }

    // C/D layout: VGPR r, lanes 0-15 -> (M=r, N=lane); lanes 16-31 -> (M=r+8).
    const int rbase = row0 + half * 8;
#pragma unroll
    for (int r = 0; r < 8; ++r) {
        Y[(size_t)(rbase + r) * GCN_H + col0 + m] = acc[r];
    }
}

// ---------- edge scatter: agg[dst] += h[src] * dinv[src]*dinv[dst] ----------
// One wave per edge; each lane moves a float4 (lane*4 .. lane*4+3 of 128).
__global__ void gcn_scatter(const float* __restrict__ h, float* __restrict__ agg,
                            const int* __restrict__ src, const int* __restrict__ dst,
                            const float* __restrict__ dinv, int E) {
    int gtid = blockIdx.x * blockDim.x + threadIdx.x;
    int e    = gtid >> 5;
    int lane = gtid & 31;
    if (e >= E) return;
    int   s   = src[e];
    int   d   = dst[e];
    float nrm = dinv[s] * dinv[d];
    const float4* hp = (const float4*)(h + (size_t)s * GCN_H);
    float4 v = hp[lane];
    float* ap = agg + (size_t)d * GCN_H + lane * 4;
    atomicAdd(ap + 0, v.x * nrm);
    atomicAdd(ap + 1, v.y * nrm);
    atomicAdd(ap + 2, v.z * nrm);
    atomicAdd(ap + 3, v.w * nrm);
}

// ---------- epilogue: out = agg + h*dinv^2 (self loop) + bias, opt. ReLU ----
__global__ void gcn_bias_act(const float* __restrict__ agg, const float* __restrict__ h,
                             const float* __restrict__ dinv, const float* __restrict__ bias,
                             float* __restrict__ out, int n_elem, int relu) {
    int idx = blockIdx.x * blockDim.x + threadIdx.x;
    if (idx >= n_elem) return;
    int   node = idx >> 7;
    int   f    = idx & 127;
    float di   = dinv[node];
    float v    = agg[idx] + h[idx] * di * di + bias[f];
    if (relu) v = fmaxf(v, 0.0f);
    out[idx] = v;
}

// ---------- pooling ----------
__global__ void gcn_pool_accum(const float* __restrict__ h, const int* __restrict__ batch,
                               float* __restrict__ pooled, float* __restrict__ cnt, int n) {
    int gtid = blockIdx.x * blockDim.x + threadIdx.x;
    int node = gtid >> 5;
    int lane = gtid & 31;
    if (node >= n) return;
    int g = batch[node];
    const float4* hp = (const float4*)(h + (size_t)node * GCN_H);
    float4 v = hp[lane];
    float* pp = pooled + (size_t)g * GCN_H + lane * 4;
    atomicAdd(pp + 0, v.x);
    atomicAdd(pp + 1, v.y);
    atomicAdd(pp + 2, v.z);
    atomicAdd(pp + 3, v.w);
    if (lane == 0) atomicAdd(cnt + g, 1.0f);
}

__global__ void gcn_pool_div(const float* __restrict__ pooled, const float* __restrict__ cnt,
                             float* __restrict__ out, int n_elem) {
    int idx = blockIdx.x * blockDim.x + threadIdx.x;
    if (idx >= n_elem) return;
    int g = idx >> 7;
    out[idx] = pooled[idx] / fmaxf(cnt[g], 1.0f);
}

// ---------------------------------------------------------------------------
extern "C" void kernel_launch(void* const* d_in, const int* in_sizes, int n_in,
                              void* d_out, int out_size, void* d_ws, size_t ws_size,
                              hipStream_t stream) {
    const float* x     = (const float*)d_in[0];   // [N,128]
    const int*   edge  = (const int*)d_in[1];     // [2,E] flat: src then dst
    const int*   batch = (const int*)d_in[2];     // [N]
    const float* W1    = (const float*)d_in[3];   // [128,128]
    const float* b1    = (const float*)d_in[4];   // [128]
    const float* W2    = (const float*)d_in[5];   // [128,128]
    const float* b2    = (const float*)d_in[6];   // [128]
    float*       out   = (float*)d_out;           // [G,128]

    const int N = in_sizes[2];
    const int E = in_sizes[1] / 2;
    const int G = out_size / GCN_H;
    const int NH = N * GCN_H;

    const int* esrc = edge;
    const int* edst = edge + E;

    // workspace carve-up (floats): bufA | bufB | dinv | deg | pooled | cnt
    float* bufA   = (float*)d_ws;
    float* bufB   = bufA + (size_t)NH;
    float* dinv   = bufB + (size_t)NH;
    float* deg    = dinv + N;
    float* pooled = deg + N;
    float* cnt    = pooled + (size_t)G * GCN_H;

    const int T = 256;
    const int nBlkN   = (N + T - 1) / T;
    const int nBlkE   = (E + T - 1) / T;
    const int nBlkNH  = (NH + T - 1) / T;
    const int nBlkEW  = (E * 32 + T - 1) / T;    // wave-per-edge
    const int nBlkNW  = (N * 32 + T - 1) / T;    // wave-per-node
    const int nBlkGEMM = N / 16;                 // 50000/16 = 3125 exactly

    // --- degree / normalization ---
    gcn_deg_init<<<nBlkN, T, 0, stream>>>(deg, N);
    gcn_deg_count<<<nBlkE, T, 0, stream>>>(edst, deg, E);
    gcn_dinv<<<nBlkN, T, 0, stream>>>(deg, dinv, N);

    // --- layer 1 ---
    gcn_gemm_f32<<<nBlkGEMM, T, 0, stream>>>(x, W1, bufA);            // h1 -> A
    gcn_zero<<<nBlkNH, T, 0, stream>>>(bufB, NH);
    gcn_scatter<<<nBlkEW, T, 0, stream>>>(bufA, bufB, esrc, edst, dinv, E);
    gcn_bias_act<<<nBlkNH, T, 0, stream>>>(bufB, bufA, dinv, b1, bufA, NH, 1); // a1 -> A

    // --- layer 2 ---
    gcn_gemm_f32<<<nBlkGEMM, T, 0, stream>>>(bufA, W2, bufB);         // h2 -> B
    gcn_zero<<<nBlkNH, T, 0, stream>>>(bufA, NH);
    gcn_scatter<<<nBlkEW, T, 0, stream>>>(bufB, bufA, esrc, edst, dinv, E);
    gcn_bias_act<<<nBlkNH, T, 0, stream>>>(bufA, bufB, dinv, b2, bufA, NH, 0); // h2' -> A

    // --- mean pool by graph ---
    gcn_zero<<<(G * GCN_H + G + T - 1) / T, T, 0, stream>>>(pooled, G * GCN_H + G);
    gcn_pool_accum<<<nBlkNW, T, 0, stream>>>(bufA, batch, pooled, cnt, N);
    gcn_pool_div<<<(G * GCN_H + T - 1) / T, T, 0, stream>>>(pooled, cnt, out, G * GCN_H);
}